// LGNetworkInterp_55353538511112
// MI455X (gfx1250) — compile-verified
//
#include <hip/hip_runtime.h>

typedef __attribute__((ext_vector_type(2))) float v2f;
typedef __attribute__((ext_vector_type(8))) float v8f;

#define BATCH 256
#define NPIX  784
#define BN    (BATCH * NPIX)     // 200704
#define PI_F  3.14159265358979323846f

// ---------------------------------------------------------------------------
// L0(z) for integer z: sum_{p=-27}^{27} -2*pi*p/56^2 * sin(2*pi*p*z/56)
// odd symmetry in p -> 2 * sum_{p=1}^{27}
// ---------------------------------------------------------------------------
__device__ __forceinline__ float L0_int(int z) {
  float acc = 0.f;
  float zf = (float)z;
  #pragma unroll
  for (int p = 1; p <= 27; ++p) {
    float fp = (float)p;
    acc -= (4.f * PI_F * fp / 3136.f) * sinf((2.f * PI_F / 56.f) * fp * zf);
  }
  return acc;
}

// ---------------------------------------------------------------------------
// h = concat([x, y], axis=-1)   (256 x 1568)
// ---------------------------------------------------------------------------
__global__ void build_h_kernel(const float* __restrict__ x,
                               const float* __restrict__ y,
                               float* __restrict__ h) {
  int idx = blockIdx.x * blockDim.x + threadIdx.x;
  if (idx >= BATCH * 2 * NPIX) return;
  int b = idx / (2 * NPIX);
  int c = idx % (2 * NPIX);
  h[idx] = (c < NPIX) ? x[b * NPIX + c] : y[b * NPIX + (c - NPIX)];
}

// ---------------------------------------------------------------------------
// Zero-padded weight copy: W (fi x fo) -> Wp (fip x fop)
// ---------------------------------------------------------------------------
__global__ void pad_w_kernel(const float* __restrict__ W, float* __restrict__ Wp,
                             int fi, int fo, int fip, int fop) {
  int idx = blockIdx.x * blockDim.x + threadIdx.x;
  if (idx >= fip * fop) return;
  int r = idx / fop;
  int c = idx % fop;
  Wp[idx] = (r < fi && c < fo) ? W[r * fo + c] : 0.f;
}

// ---------------------------------------------------------------------------
// Gt[j*784 + i] = G[i,j] = a_i * D0[i,j] + b_i * D1[i,j]
//   pixel i: row ri = i/28, col ci = i%28; coord (cx,cy) = (ri-14, ci-14)
//   D0[i,j] = L0(ri - rj) iff ci == cj ; D1[i,j] = L0(ci - cj) iff ri == rj
//   a_i = P[0,0] + P[0,1]*cx + P[0,2]*cy ; b_i = P[1,0] + P[1,1]*cx + P[1,2]*cy
// ---------------------------------------------------------------------------
__global__ void build_Gt_kernel(const float* __restrict__ P, float* __restrict__ Gt) {
  int idx = blockIdx.x * blockDim.x + threadIdx.x;
  if (idx >= NPIX * NPIX) return;
  int j = idx / NPIX;
  int i = idx % NPIX;
  int ri = i / 28, ci = i % 28;
  int rj = j / 28, cj = j % 28;
  float cx = (float)(ri - 14), cy = (float)(ci - 14);
  float g = 0.f;
  if (ci == cj && ri != rj) {
    float a = P[0] + P[1] * cx + P[2] * cy;
    g += a * L0_int(ri - rj);
  }
  if (ri == rj && ci != cj) {
    float bb = P[3] + P[4] * cx + P[5] * cy;
    g += bb * L0_int(ci - cj);
  }
  Gt[idx] = g;
}

// ---------------------------------------------------------------------------
// f32 GEMM on V_WMMA_F32_16X16X4_F32, 64(M) x 16(N) C block per wave32.
// C(MxN) = A(MxK) @ B(KxN) [+ bias, + relu], row-major, M % 64 == 0,
// N % 16 == 0, K % 4 == 0. The B fragment (strided loads) is loaded once per
// K-step and reused by 4 WMMAs -> 320 B fragment traffic per 2048-FLOP WMMA.
// Tile counts are always % 4 -> every wave fully active (EXEC all ones).
// ISA layouts (§7.12.2):
//   A  16x4 : lane -> M = lane%16, VGPR v -> K = 2*(lane/16) + v
//   B  4x16 : lane -> N = lane%16, VGPR v -> K = 2*(lane/16) + v
//   C 16x16 : lane -> N = lane%16, VGPR v -> M = v + 8*(lane/16)
// ---------------------------------------------------------------------------
__global__ void gemm_wmma_f32(const float* __restrict__ A,
                              const float* __restrict__ Bm,
                              const float* __restrict__ bias,
                              float* __restrict__ C,
                              int M, int N, int K, int Nreal, int relu) {
  int lane = threadIdx.x & 31;
  int wave = threadIdx.x >> 5;
  int tilesN = N >> 4;
  int tiles  = (M >> 6) * tilesN;
  int tile   = blockIdx.x * (blockDim.x >> 5) + wave;
  if (tile >= tiles) return;                 // wave-uniform; grids are exact
  int mBase = (tile / tilesN) << 6;
  int nBase = (tile % tilesN) << 4;
  int half = lane >> 4;
  int l16  = lane & 15;

  v8f acc0 = {}, acc1 = {}, acc2 = {}, acc3 = {};
  const float* a0 = A + (size_t)(mBase + l16) * K + 2 * half;
  const float* a1 = a0 + (size_t)16 * K;
  const float* a2 = a0 + (size_t)32 * K;
  const float* a3 = a0 + (size_t)48 * K;
  const float* bcol = Bm + nBase + l16;

  for (int k0 = 0; k0 < K; k0 += 4) {
    int kb = k0 + 2 * half;
    v2f b;
    b.x = bcol[(size_t)kb * N];
    b.y = bcol[(size_t)(kb + 1) * N];
    v2f a;
    a.x = a0[k0]; a.y = a0[k0 + 1];
    acc0 = __builtin_amdgcn_wmma_f32_16x16x4_f32(false, a, false, b, (short)0,
                                                 acc0, false, false);
    a.x = a1[k0]; a.y = a1[k0 + 1];
    acc1 = __builtin_amdgcn_wmma_f32_16x16x4_f32(false, a, false, b, (short)0,
                                                 acc1, false, false);
    a.x = a2[k0]; a.y = a2[k0 + 1];
    acc2 = __builtin_amdgcn_wmma_f32_16x16x4_f32(false, a, false, b, (short)0,
                                                 acc2, false, false);
    a.x = a3[k0]; a.y = a3[k0 + 1];
    acc3 = __builtin_amdgcn_wmma_f32_16x16x4_f32(false, a, false, b, (short)0,
                                                 acc3, false, false);
  }

  int col = nBase + l16;
  float bv = (bias != nullptr && col < Nreal) ? bias[col] : 0.f;
  v8f accs[4] = {acc0, acc1, acc2, acc3};
  #pragma unroll
  for (int t = 0; t < 4; ++t) {
    #pragma unroll
    for (int v = 0; v < 8; ++v) {
      int row = mBase + 16 * t + v + 8 * half;
      float val = accs[t][v] + bv;
      if (relu) val = fmaxf(val, 0.f);
      C[(size_t)row * N + col] = val;
    }
  }
}

// ---------------------------------------------------------------------------
// y_hat[b,i] = sum_k theta_b^k / k! * V_k[b,i]  (V_0 = x, V_1..V_6 in ws)
// Also emits the pass-through tuple members: y, theta, theta_true, x.
// ---------------------------------------------------------------------------
__global__ void combine_kernel(const float* __restrict__ x,
                               const float* __restrict__ y,
                               const float* __restrict__ theta_true,
                               const float* __restrict__ actF,   // 256 x 16
                               const float* __restrict__ V,      // 6 * BN
                               float* __restrict__ out) {
  int idx = blockIdx.x * blockDim.x + threadIdx.x;
  if (idx >= BN) return;
  int b = idx / NPIX;
  float th = actF[b * 16];
  const float invfac[7] = {1.f, 1.f, 0.5f, 1.f / 6.f, 1.f / 24.f,
                           1.f / 120.f, 1.f / 720.f};
  float acc = x[idx];           // k = 0 term: coeff = 1
  float tp = 1.f;
  #pragma unroll
  for (int k = 1; k <= 6; ++k) {
    tp *= th;
    acc += tp * invfac[k] * V[(size_t)(k - 1) * BN + idx];
  }
  out[idx] = acc;               // y_hat
  out[BN + idx] = y[idx];       // y
  out[2 * BN + 512 + idx] = x[idx];  // x (after theta, theta_true)
  if (idx < BATCH) {
    out[2 * BN + idx] = actF[idx * 16];        // theta
    out[2 * BN + BATCH + idx] = theta_true[idx];
  }
}

// ---------------------------------------------------------------------------
extern "C" void kernel_launch(void* const* d_in, const int* in_sizes, int n_in,
                              void* d_out, int out_size, void* d_ws, size_t ws_size,
                              hipStream_t stream) {
  const float* x          = (const float*)d_in[0];
  const float* y          = (const float*)d_in[1];
  const float* theta_true = (const float*)d_in[2];
  const float* W[7];
  const float* bvec[7];
  for (int l = 0; l < 7; ++l) {
    W[l]    = (const float*)d_in[3 + 2 * l];
    bvec[l] = (const float*)d_in[4 + 2 * l];
  }
  const float* P = (const float*)d_in[n_in - 1];

  // layer dims: real and padded-to-16
  static const int fi_r[7] = {1568, 1000, 900, 800, 500, 100, 20};
  static const int fo_r[7] = {1000, 900, 800, 500, 100, 20, 1};
  static const int fi_p[7] = {1568, 1008, 912, 800, 512, 112, 32};
  static const int fo_p[7] = {1008, 912, 800, 512, 112, 32, 16};

  float* ws = (float*)d_ws;
  size_t off = 0;
  float* actA = ws + off; off += (size_t)BATCH * 1568;
  float* actB = ws + off; off += (size_t)BATCH * 1008;
  float* Wp[7];
  for (int l = 0; l < 7; ++l) { Wp[l] = ws + off; off += (size_t)fi_p[l] * fo_p[l]; }
  float* Gt = ws + off; off += (size_t)NPIX * NPIX;
  float* V  = ws + off; off += (size_t)6 * BN;   // V_1..V_6

  // 1) h = [x|y]
  {
    int n = BATCH * 2 * NPIX;
    build_h_kernel<<<(n + 255) / 256, 256, 0, stream>>>(x, y, actA);
  }
  // 2) pad weights
  for (int l = 0; l < 7; ++l) {
    int n = fi_p[l] * fo_p[l];
    pad_w_kernel<<<(n + 255) / 256, 256, 0, stream>>>(W[l], Wp[l], fi_r[l], fo_r[l],
                                                      fi_p[l], fo_p[l]);
  }
  // 3) build G^T
  {
    int n = NPIX * NPIX;
    build_Gt_kernel<<<(n + 255) / 256, 256, 0, stream>>>(P, Gt);
  }
  // 4) MLP chain (ping-pong actA/actB); final theta lands in actB (256 x 16)
  {
    float* cur = actA;
    float* nxt = actB;
    for (int l = 0; l < 7; ++l) {
      int Mt = BATCH, Nt = fo_p[l], Kt = fi_p[l];
      int tiles = (Mt >> 6) * (Nt >> 4);      // 4*(N/16), always % 4 == 0
      gemm_wmma_f32<<<tiles / 4, 128, 0, stream>>>(cur, Wp[l], bvec[l], nxt,
                                                   Mt, Nt, Kt, fo_r[l],
                                                   (l < 6) ? 1 : 0);
      float* t = cur; cur = nxt; nxt = t;
    }
  }
  // 5) V_k = V_{k-1} @ G^T, V_0 = x   (6 GEMMs, 256x784 @ 784x784)
  {
    int tiles = (BATCH >> 6) * (NPIX >> 4);   // 4*49 = 196, % 4 == 0
    gemm_wmma_f32<<<tiles / 4, 128, 0, stream>>>(x, Gt, nullptr, V,
                                                 BATCH, NPIX, NPIX, NPIX, 0);
    for (int k = 2; k <= 6; ++k) {
      gemm_wmma_f32<<<tiles / 4, 128, 0, stream>>>(V + (size_t)(k - 2) * BN, Gt,
                                                   nullptr, V + (size_t)(k - 1) * BN,
                                                   BATCH, NPIX, NPIX, NPIX, 0);
    }
  }
  // 6) combine + pass-throughs (theta is in actB after 7 swaps)
  {
    combine_kernel<<<(BN + 255) / 256, 256, 0, stream>>>(x, y, theta_true, actB, V,
                                                         (float*)d_out);
  }
}